// ZoneGAT_47493748359412
// MI455X (gfx1250) — compile-verified
//
#include <hip/hip_runtime.h>
#include <hip/hip_bf16.h>
#include <math.h>

// ---------------------------------------------------------------------------
// GAT (3 layers) for MI455X / gfx1250.
// Dense transforms use V_WMMA_F32_16X16X4_F32 (full-precision fp32 WMMA) with
// a guard-free inner loop (inputs padded so K % 4 == 0; load rows clamped).
// Edge softmax/aggregation are fp32 atomic scatter/gather (memory bound:
// ~2.6 GB irregular traffic -> ~115 us at 23.3 TB/s; GEMMs ~1.8 GFLOP).
// ---------------------------------------------------------------------------

typedef __attribute__((ext_vector_type(2))) float v2f;
typedef __attribute__((ext_vector_type(8))) float v8f;

#define NEG_SLOPE 0.2f
#define GAT_EPS   1e-16f

// ----------------------------- WMMA fp32 GEMM ------------------------------
// Y[nrows x ncols] = X[nrows x kdim] @ W[kdim x ncols]
// Requires kdim % 4 == 0, ncols % 16 == 0. One wave = one 16x16 tile.
// A layout (16x4 f32): lanes 0-15 -> rows m, K=0,1 in v0,v1;
//                      lanes 16-31 -> rows m, K=2,3.
// B layout (4x16 f32): mirrored. C/D: VGPR i = row m0+i (lanes 0-15) /
// m0+8+i (lanes 16-31), column = lane & 15.
__global__ __launch_bounds__(32) void gat_gemm_wmma(
    const float* __restrict__ X, const float* __restrict__ W,
    float* __restrict__ Y, int nrows, int kdim, int ncols)
{
    const int lane = threadIdx.x & 31;
    const int m0   = blockIdx.x * 16;
    const int n0   = blockIdx.y * 16;
    const int half = lane >> 4;     // 0: K pair (0,1), 1: K pair (2,3)
    const int l15  = lane & 15;
    const int col  = n0 + l15;

    // Clamp load row: duplicate rows feed tile rows that are never stored.
    int row = m0 + l15;
    if (row > nrows - 1) row = nrows - 1;

    const float* __restrict__ arow = X + (size_t)row * kdim + half * 2;
    const float* __restrict__ bcol = W + (size_t)(half * 2) * ncols + col;

    v8f c = {};
    #pragma unroll 4
    for (int k = 0; k < kdim; k += 4) {
        v2f a = *(const v2f*)(arow + k);
        v2f b;
        b.x = bcol[(size_t)k * ncols];
        b.y = bcol[(size_t)k * ncols + ncols];
        c = __builtin_amdgcn_wmma_f32_16x16x4_f32(
                false, a, false, b, (short)0, c, false, false);
    }
    #pragma unroll
    for (int i = 0; i < 8; ++i) {
        const int r = m0 + half * 8 + i;
        if (r < nrows) Y[(size_t)r * ncols + col] = c[i];
    }
}

// ------------------------- padding helpers (layer 1) -----------------------
// xp[N,8] = x[N,7] zero-padded; W1p[8,128] = W1[7,128] zero-padded.
__global__ void gat_pad_x(const float* __restrict__ x, float* __restrict__ xp,
                          int N)
{
    int idx = blockIdx.x * blockDim.x + threadIdx.x;
    if (idx >= N * 8) return;
    int n = idx >> 3, k = idx & 7;
    xp[idx] = (k < 7) ? x[n * 7 + k] : 0.0f;
}

__global__ void gat_pad_w1(const float* __restrict__ w, float* __restrict__ wp)
{
    int idx = blockIdx.x * blockDim.x + threadIdx.x;
    if (idx >= 8 * 128) return;
    int k = idx >> 7, f = idx & 127;
    wp[idx] = (k < 7) ? w[k * 128 + f] : 0.0f;
}

// ------------------------- attention logits --------------------------------
// alS[n,h] = sum_c h[n,h*C+c]*a_src[h*C+c]; same for alD.
template <int H, int C>
__global__ void gat_logits(const float* __restrict__ h,
                           const float* __restrict__ a_src,
                           const float* __restrict__ a_dst,
                           float* __restrict__ alS, float* __restrict__ alD,
                           int N)
{
    int idx = blockIdx.x * blockDim.x + threadIdx.x;   // node*H + head
    if (idx >= N * H) return;
    int n  = idx / H;
    int hh = idx - n * H;
    const float* row = h + (size_t)n * (H * C) + hh * C;
    const float* as  = a_src + hh * C;
    const float* ad  = a_dst + hh * C;
    float s1 = 0.0f, s2 = 0.0f;
    #pragma unroll
    for (int c = 0; c < C; ++c) {
        float v = row[c];
        s1 += v * as[c];
        s2 += v * ad[c];
    }
    alS[idx] = s1;
    alD[idx] = s2;
}

// ------------------------------ fill ---------------------------------------
__global__ void gat_fill(float* __restrict__ p, float v, int n)
{
    int i = blockIdx.x * blockDim.x + threadIdx.x;
    if (i < n) p[i] = v;
}

// -------------------- float atomic max (monotone encoding) -----------------
__device__ __forceinline__ void atomicMaxF(float* addr, float val)
{
    if (val >= 0.0f) atomicMax((int*)addr, __float_as_int(val));
    else             atomicMin((unsigned int*)addr, __float_as_uint(val));
}

__device__ __forceinline__ float leaky(float v)
{
    return v > 0.0f ? v : NEG_SLOPE * v;
}

// edges e in [0,E) come from edge_index; e in [E, E+N) are self loops.
// ei layout: [0..E) = src row, [E..2E) = dst row.
__device__ __forceinline__ void edge_nodes(const int* __restrict__ ei,
                                           int e, int E, int& src, int& dst)
{
    if (e < E) { src = ei[e]; dst = ei[E + e]; }
    else       { src = e - E; dst = e - E; }
}

// ------------------------- pass 1: segment max -----------------------------
template <int H>
__global__ void gat_edge_max(const int* __restrict__ ei, int E, int N,
                             const float* __restrict__ alS,
                             const float* __restrict__ alD,
                             float* __restrict__ m)
{
    int t = blockIdx.x * blockDim.x + threadIdx.x;
    int total = (E + N) * H;
    if (t >= total) return;
    int e  = t / H;
    int hh = t - e * H;
    int src, dst;
    edge_nodes(ei, e, E, src, dst);
    float v = leaky(alS[src * H + hh] + alD[dst * H + hh]);
    atomicMaxF(&m[dst * H + hh], v);
}

// ------------------------- pass 2: segment sum of exp ----------------------
template <int H>
__global__ void gat_edge_sum(const int* __restrict__ ei, int E, int N,
                             const float* __restrict__ alS,
                             const float* __restrict__ alD,
                             const float* __restrict__ m,
                             float* __restrict__ s)
{
    int t = blockIdx.x * blockDim.x + threadIdx.x;
    int total = (E + N) * H;
    if (t >= total) return;
    int e  = t / H;
    int hh = t - e * H;
    int src, dst;
    edge_nodes(ei, e, E, src, dst);
    float v  = leaky(alS[src * H + hh] + alD[dst * H + hh]);
    float ex = expf(v - m[dst * H + hh]);
    atomicAdd(&s[dst * H + hh], ex);
}

// ------------------- pass 3: weighted scatter aggregation ------------------
// One thread per (edge, head, channel) -> coalesced h[src] reads and
// coalesced atomic adds on out[dst] rows. H, C compile-time so index math
// lowers to shifts/mul-by-constant (no integer division).
template <int H, int C>
__global__ void gat_edge_aggr(const int* __restrict__ ei, int E, int N,
                              const float* __restrict__ alS,
                              const float* __restrict__ alD,
                              const float* __restrict__ m,
                              const float* __restrict__ s,
                              const float* __restrict__ h,
                              float* __restrict__ out)
{
    constexpr int HC = H * C;
    long long idx = (long long)blockIdx.x * blockDim.x + threadIdx.x;
    long long total = (long long)(E + N) * HC;
    if (idx >= total) return;
    int e  = (int)(idx / HC);
    int r  = (int)(idx - (long long)e * HC);
    int hh = r / C;
    int src, dst;
    edge_nodes(ei, e, E, src, dst);
    float v     = leaky(alS[src * H + hh] + alD[dst * H + hh]);
    float ex    = expf(v - m[dst * H + hh]);
    float alpha = ex / (s[dst * H + hh] + GAT_EPS);
    atomicAdd(&out[(size_t)dst * HC + r], h[(size_t)src * HC + r] * alpha);
}

// --------------------------- bias + ELU ------------------------------------
__global__ void gat_bias_elu(float* __restrict__ y,
                             const float* __restrict__ b, int N, int F)
{
    int idx = blockIdx.x * blockDim.x + threadIdx.x;
    if (idx >= N * F) return;
    int f = idx & (F - 1);                     // F is 128 or 64 (power of 2)
    float v = y[idx] + b[f];
    y[idx] = v > 0.0f ? v : (expf(v) - 1.0f);  // jax.nn.elu
}

// -------------------- layer 3 transform (64 -> 1) + logits -----------------
__global__ void gat_h3(const float* __restrict__ x3,
                       const float* __restrict__ W3,
                       const float* __restrict__ as3,
                       const float* __restrict__ ad3,
                       float* __restrict__ h3,
                       float* __restrict__ alS, float* __restrict__ alD, int N)
{
    int n = blockIdx.x * blockDim.x + threadIdx.x;
    if (n >= N) return;
    const float* row = x3 + (size_t)n * 64;
    float acc = 0.0f;
    #pragma unroll 8
    for (int k = 0; k < 64; ++k) acc += row[k] * W3[k];
    h3[n]  = acc;
    alS[n] = acc * as3[0];
    alD[n] = acc * ad3[0];
}

// -------------------------- final sigmoid * 100 ----------------------------
__global__ void gat_final(const float* __restrict__ agg,
                          const float* __restrict__ b3,
                          float* __restrict__ out, int N)
{
    int n = blockIdx.x * blockDim.x + threadIdx.x;
    if (n >= N) return;
    float v = agg[n] + b3[0];
    out[n] = 100.0f / (1.0f + expf(-v));
}

// ---------------------------------------------------------------------------
extern "C" void kernel_launch(void* const* d_in, const int* in_sizes, int n_in,
                              void* d_out, int out_size, void* d_ws, size_t ws_size,
                              hipStream_t stream)
{
    const float* x   = (const float*)d_in[0];
    const int*   ei  = (const int*)  d_in[1];
    const float* W1  = (const float*)d_in[2];
    const float* as1 = (const float*)d_in[3];
    const float* ad1 = (const float*)d_in[4];
    const float* b1  = (const float*)d_in[5];
    const float* W2  = (const float*)d_in[6];
    const float* as2 = (const float*)d_in[7];
    const float* ad2 = (const float*)d_in[8];
    const float* b2  = (const float*)d_in[9];
    const float* W3  = (const float*)d_in[10];
    const float* as3 = (const float*)d_in[11];
    const float* ad3 = (const float*)d_in[12];
    const float* b3  = (const float*)d_in[13];
    float* out = (float*)d_out;

    const int N  = in_sizes[0] / 7;
    const int E  = in_sizes[1] / 2;
    const int ET = E + N;                       // edges incl. self loops

    // workspace layout (floats)
    float* ws   = (float*)d_ws;
    float* bufA = ws;                           // N*128 : h (per layer)
    float* bufB = bufA + (size_t)N * 128;       // N*128 : out / next input
    float* alS  = bufB + (size_t)N * 128;       // N*4
    float* alD  = alS  + (size_t)N * 4;         // N*4
    float* mB   = alD  + (size_t)N * 4;         // N*4
    float* sB   = mB   + (size_t)N * 4;         // N*4
    float* xp   = sB   + (size_t)N * 4;         // N*8  : padded x
    float* W1p  = xp   + (size_t)N * 8;         // 8*128: padded W1

    const float NEG_INF = __builtin_bit_cast(float, (unsigned)0xFF800000u);
    const int TB = 256;
    auto blk = [](long long n, int tb) { return (unsigned)((n + tb - 1) / tb); };

    // =================== Layer 1: 7 -> 128 (H=4, C=32) =====================
    {
        const int F = 128;
        gat_pad_x<<<blk((long long)N * 8, TB), TB, 0, stream>>>(x, xp, N);
        gat_pad_w1<<<blk(8 * 128, TB), TB, 0, stream>>>(W1, W1p);
        dim3 g1((N + 15) / 16, F / 16);
        gat_gemm_wmma<<<g1, 32, 0, stream>>>(xp, W1p, bufA, N, 8, F);
        gat_logits<4, 32><<<blk((long long)N * 4, TB), TB, 0, stream>>>(
            bufA, as1, ad1, alS, alD, N);
        gat_fill<<<blk((long long)N * 4, TB), TB, 0, stream>>>(mB, NEG_INF, N * 4);
        gat_fill<<<blk((long long)N * 4, TB), TB, 0, stream>>>(sB, 0.0f, N * 4);
        gat_fill<<<blk((long long)N * F, TB), TB, 0, stream>>>(bufB, 0.0f, N * F);
        gat_edge_max<4><<<blk((long long)ET * 4, TB), TB, 0, stream>>>(
            ei, E, N, alS, alD, mB);
        gat_edge_sum<4><<<blk((long long)ET * 4, TB), TB, 0, stream>>>(
            ei, E, N, alS, alD, mB, sB);
        gat_edge_aggr<4, 32><<<blk((long long)ET * 128, TB), TB, 0, stream>>>(
            ei, E, N, alS, alD, mB, sB, bufA, bufB);
        gat_bias_elu<<<blk((long long)N * F, TB), TB, 0, stream>>>(bufB, b1, N, F);
    }

    // =================== Layer 2: 128 -> 64 (H=4, C=16) ====================
    {
        const int F = 64;
        dim3 g2((N + 15) / 16, F / 16);
        gat_gemm_wmma<<<g2, 32, 0, stream>>>(bufB, W2, bufA, N, 128, F);
        gat_logits<4, 16><<<blk((long long)N * 4, TB), TB, 0, stream>>>(
            bufA, as2, ad2, alS, alD, N);
        gat_fill<<<blk((long long)N * 4, TB), TB, 0, stream>>>(mB, NEG_INF, N * 4);
        gat_fill<<<blk((long long)N * 4, TB), TB, 0, stream>>>(sB, 0.0f, N * 4);
        gat_fill<<<blk((long long)N * F, TB), TB, 0, stream>>>(bufB, 0.0f, N * F);
        gat_edge_max<4><<<blk((long long)ET * 4, TB), TB, 0, stream>>>(
            ei, E, N, alS, alD, mB);
        gat_edge_sum<4><<<blk((long long)ET * 4, TB), TB, 0, stream>>>(
            ei, E, N, alS, alD, mB, sB);
        gat_edge_aggr<4, 16><<<blk((long long)ET * 64, TB), TB, 0, stream>>>(
            ei, E, N, alS, alD, mB, sB, bufA, bufB);
        gat_bias_elu<<<blk((long long)N * F, TB), TB, 0, stream>>>(bufB, b2, N, F);
    }

    // =================== Layer 3: 64 -> 1 (H=1, C=1) =======================
    {
        float* h3   = bufA;           // N
        float* agg3 = bufA + N;       // N
        gat_h3<<<blk(N, TB), TB, 0, stream>>>(bufB, W3, as3, ad3, h3, alS, alD, N);
        gat_fill<<<blk(N, TB), TB, 0, stream>>>(mB, NEG_INF, N);
        gat_fill<<<blk(N, TB), TB, 0, stream>>>(sB, 0.0f, N);
        gat_fill<<<blk(N, TB), TB, 0, stream>>>(agg3, 0.0f, N);
        gat_edge_max<1><<<blk((long long)ET, TB), TB, 0, stream>>>(
            ei, E, N, alS, alD, mB);
        gat_edge_sum<1><<<blk((long long)ET, TB), TB, 0, stream>>>(
            ei, E, N, alS, alD, mB, sB);
        gat_edge_aggr<1, 1><<<blk((long long)ET, TB), TB, 0, stream>>>(
            ei, E, N, alS, alD, mB, sB, h3, agg3);
        gat_final<<<blk(N, TB), TB, 0, stream>>>(agg3, b3, out, N);
    }
}